// Graph_Transformer_39848706572459
// MI455X (gfx1250) — compile-verified
//
#include <hip/hip_runtime.h>
#include <hip/hip_bf16.h>

// ---------------------------------------------------------------------------
// Graph Transformer block for MI455X (gfx1250, wave32, WMMA).
// N=50000 nodes, DM=128, H=8, HD=16, FF=512, E=640000 edges.
// fp32 throughout; dense math on V_WMMA_F32_16X16X4_F32, register-blocked so
// the inner loops are WMMA/DS-issue bound; all GEMM strides are compile-time
// so B loads use immediate offsets off a single base address.
// ---------------------------------------------------------------------------

typedef float v2f __attribute__((ext_vector_type(2)));
typedef float v8f __attribute__((ext_vector_type(8)));

__device__ __forceinline__ v8f wmma_f32x4(v2f a, v2f b, v8f c) {
  // D(16x16,f32) = A(16x4,f32) * B(4x16,f32) + C
  return __builtin_amdgcn_wmma_f32_16x16x4_f32(false, a, false, b, (short)0, c,
                                               false, false);
}

__device__ __forceinline__ float waveReduceSum(float v) {
  #pragma unroll
  for (int off = 16; off > 0; off >>= 1) v += __shfl_xor(v, off, 32);
  return v;
}

__device__ __forceinline__ void atomicMaxF(float* addr, float val) {
  int old = __float_as_int(*addr);
  while (__int_as_float(old) < val) {
    int assumed = old;
    old = atomicCAS((int*)addr, assumed, __float_as_int(val));
    if (old == assumed) break;
  }
}

// ---------------------------------------------------------------------------
// Pack Wq|Wk|Wv|Wskip (each 128x128 row-major) -> Wcat[128][512], bcat[512].
// ---------------------------------------------------------------------------
__global__ void pack_w_kernel(const float* __restrict__ Wq, const float* __restrict__ Wk,
                              const float* __restrict__ Wv, const float* __restrict__ Wsk,
                              const float* __restrict__ bq, const float* __restrict__ bk,
                              const float* __restrict__ bv, const float* __restrict__ bsk,
                              float* __restrict__ Wcat, float* __restrict__ bcat) {
  int idx = blockIdx.x * 256 + threadIdx.x;
  if (idx < 128 * 512) {
    int k = idx >> 9;        // row of Wcat
    int c = idx & 511;       // col of Wcat
    int s = c >> 7;          // which weight matrix
    int j = c & 127;
    const float* W = (s == 0) ? Wq : (s == 1) ? Wk : (s == 2) ? Wv : Wsk;
    Wcat[idx] = W[k * 128 + j];
  }
  if (idx < 512) {
    int s = idx >> 7, j = idx & 127;
    const float* b = (s == 0) ? bq : (s == 1) ? bk : (s == 2) ? bv : bsk;
    bcat[idx] = b[j];
  }
}

// ---------------------------------------------------------------------------
// K=128 WMMA GEMM:  Y[M,NC] = X[M,128] @ W[128,NC] + bias   (NC compile-time)
// Block = 256 threads (8 waves). Block tile = 64 rows x 128 cols.
// Each wave owns one 16-col stripe over 4 row tiles:
//   - B column fragment for all of K (32 x v2f = 64 VGPRs) preloaded to
//     registers once (immediate-offset loads), reused for 128 WMMAs.
//   - A fragments from LDS (ds_load_2addr_b64).
// ---------------------------------------------------------------------------
template <int NC>
__global__ __launch_bounds__(256) void gemm128_bias_kernel(
    const float* __restrict__ X, const float* __restrict__ W,
    const float* __restrict__ bias, float* __restrict__ Y, int M) {
  __shared__ float Xs[64][132];  // 33.8 KB; 132-float stride keeps rows 16B aligned

  const int tid = threadIdx.x;
  const int wave = tid >> 5, lane = tid & 31;
  const int rowBase = blockIdx.x * 64;
  const int colBase = blockIdx.y * 128;

  // Stage X tile: 64 rows x 128 cols, 32 floats per thread (clamped rows keep
  // EXEC full for the WMMAs below).
  {
    int r = tid >> 2;            // 0..63
    int c0 = (tid & 3) * 32;     // 0,32,64,96
    int rr = rowBase + r;
    if (rr > M - 1) rr = M - 1;
    const float* src = X + (size_t)rr * 128 + c0;
    #pragma unroll
    for (int i = 0; i < 8; ++i) {
      float4 t = *(const float4*)(src + 4 * i);
      Xs[r][c0 + 4 * i + 0] = t.x;
      Xs[r][c0 + 4 * i + 1] = t.y;
      Xs[r][c0 + 4 * i + 2] = t.z;
      Xs[r][c0 + 4 * i + 3] = t.w;
    }
  }
  __syncthreads();

  const int m = lane & 15;
  const int kb_off = (lane >> 4) << 1;  // 0 or 2
  const int hi8 = (lane >> 4) << 3;     // 0 or 8
  const int col = colBase + wave * 16 + m;

  // Preload B fragments for the full K dimension into registers.
  // kb is compile-time, NC is compile-time -> single base + immediate offsets.
  const float* Wb = W + col + (size_t)kb_off * NC;
  v2f Bf[32];
  #pragma unroll
  for (int i = 0; i < 32; ++i) {
    Bf[i].x = Wb[(size_t)(4 * i) * NC];
    Bf[i].y = Wb[(size_t)(4 * i + 1) * NC];
  }

  v8f acc0 = {}, acc1 = {}, acc2 = {}, acc3 = {};
  #pragma unroll
  for (int i = 0; i < 32; ++i) {
    int kb = 4 * i + kb_off;
    v2f a0; a0.x = Xs[m][kb];      a0.y = Xs[m][kb + 1];
    v2f a1; a1.x = Xs[16 + m][kb]; a1.y = Xs[16 + m][kb + 1];
    v2f a2; a2.x = Xs[32 + m][kb]; a2.y = Xs[32 + m][kb + 1];
    v2f a3; a3.x = Xs[48 + m][kb]; a3.y = Xs[48 + m][kb + 1];
    acc0 = wmma_f32x4(a0, Bf[i], acc0);
    acc1 = wmma_f32x4(a1, Bf[i], acc1);
    acc2 = wmma_f32x4(a2, Bf[i], acc2);
    acc3 = wmma_f32x4(a3, Bf[i], acc3);
  }

  const float bv = bias[col];
  float* Yb = Y + (size_t)(rowBase + hi8) * NC + col;
  #pragma unroll
  for (int j = 0; j < 8; ++j) {
    int r0 = rowBase + j + hi8;
    if (r0 < M)      Yb[(size_t)j * NC]        = acc0[j] + bv;
    if (r0 + 16 < M) Yb[(size_t)(j + 16) * NC] = acc1[j] + bv;
    if (r0 + 32 < M) Yb[(size_t)(j + 32) * NC] = acc2[j] + bv;
    if (r0 + 48 < M) Yb[(size_t)(j + 48) * NC] = acc3[j] + bv;
  }
}

// ---------------------------------------------------------------------------
// Init: maxa=-inf, den=0, conv = skip part of qkvs (x@Wskip+bskip).
// ---------------------------------------------------------------------------
__global__ void init_buffers_kernel(float* __restrict__ maxa, float* __restrict__ den,
                                    const float* __restrict__ qkvs,
                                    float* __restrict__ conv, int N) {
  int i = blockIdx.x * 256 + threadIdx.x;
  if (i < N * 8) { maxa[i] = -3.0e38f; den[i] = 0.0f; }
  if (i < N * 128) {
    int n = i >> 7, j = i & 127;
    conv[i] = qkvs[(size_t)n * 512 + 384 + j];
  }
}

// ---------------------------------------------------------------------------
// Edge logits: alpha[e,h] = dot(q[dst,h], k[src,h]) / 4 ; atomic segment max.
// One thread per (edge, head). All gathers hit the 192 MB L2 (q/k arrays are
// 25.6 MB each).
// ---------------------------------------------------------------------------
__global__ void edge_logits_kernel(const float* __restrict__ qkvs,
                                   const long long* __restrict__ ei,
                                   float* __restrict__ alpha, float* __restrict__ maxa,
                                   int E) {
  int idx = blockIdx.x * 256 + threadIdx.x;
  if (idx >= E * 8) return;
  int e = idx >> 3, h = idx & 7;
  long long s = ei[e];       // src
  long long d = ei[E + e];   // dst
  const float4* q = (const float4*)(qkvs + (size_t)d * 512 + 0 + h * 16);
  const float4* k = (const float4*)(qkvs + (size_t)s * 512 + 128 + h * 16);
  float dot = 0.0f;
  #pragma unroll
  for (int i = 0; i < 4; ++i) {
    float4 qa = q[i], ka = k[i];
    dot += qa.x * ka.x + qa.y * ka.y + qa.z * ka.z + qa.w * ka.w;
  }
  float a = dot * 0.25f;  // 1/sqrt(16)
  alpha[idx] = a;
  atomicMaxF(maxa + (size_t)d * 8 + h, a);
}

// ---------------------------------------------------------------------------
// exp(alpha - max) in place; atomic segment sum into den.
// ---------------------------------------------------------------------------
__global__ void edge_expden_kernel(float* __restrict__ alpha,
                                   const long long* __restrict__ ei,
                                   const float* __restrict__ maxa,
                                   float* __restrict__ den, int E) {
  int idx = blockIdx.x * 256 + threadIdx.x;
  if (idx >= E * 8) return;
  int e = idx >> 3, h = idx & 7;
  long long d = ei[E + e];
  float ex = expf(alpha[idx] - maxa[(size_t)d * 8 + h]);
  alpha[idx] = ex;
  atomicAdd(den + (size_t)d * 8 + h, ex);
}

// ---------------------------------------------------------------------------
// Aggregate: conv[dst] += (ex/den) * v[src]. One wave (32 lanes) per edge;
// lane L covers feature cols 4L..4L+3 (head h = L>>2).
// ---------------------------------------------------------------------------
__global__ void edge_aggregate_kernel(const float* __restrict__ qkvs,
                                      const long long* __restrict__ ei,
                                      const float* __restrict__ exv,
                                      const float* __restrict__ den,
                                      float* __restrict__ conv, int E) {
  int gw = (blockIdx.x * 256 + threadIdx.x) >> 5;
  int L = threadIdx.x & 31;
  if (gw >= E) return;
  long long s = ei[gw];
  long long d = ei[E + gw];
  int h = L >> 2;
  float a = exv[(size_t)gw * 8 + h] / (den[(size_t)d * 8 + h] + 1e-16f);
  float4 vv = *(const float4*)(qkvs + (size_t)s * 512 + 256 + L * 4);
  float* p = conv + (size_t)d * 128 + L * 4;
  atomicAdd(p + 0, a * vv.x);
  atomicAdd(p + 1, a * vv.y);
  atomicAdd(p + 2, a * vv.z);
  atomicAdd(p + 3, a * vv.w);
}

// ---------------------------------------------------------------------------
// LayerNorm(residual): Y = LN(A + B) * g + b.  One wave per row (128 cols).
// ---------------------------------------------------------------------------
__global__ __launch_bounds__(256) void ln_residual_kernel(
    const float* __restrict__ A, const float* __restrict__ B,
    const float* __restrict__ g, const float* __restrict__ b,
    float* __restrict__ Y, int M) {
  int row = blockIdx.x * 8 + (threadIdx.x >> 5);
  int L = threadIdx.x & 31;
  if (row >= M) return;
  float4 av = *(const float4*)(A + (size_t)row * 128 + L * 4);
  float4 rv = *(const float4*)(B + (size_t)row * 128 + L * 4);
  float t0 = av.x + rv.x, t1 = av.y + rv.y, t2 = av.z + rv.z, t3 = av.w + rv.w;
  float mu = waveReduceSum(t0 + t1 + t2 + t3) * (1.0f / 128.0f);
  float d0 = t0 - mu, d1 = t1 - mu, d2 = t2 - mu, d3 = t3 - mu;
  float var = waveReduceSum(d0 * d0 + d1 * d1 + d2 * d2 + d3 * d3) * (1.0f / 128.0f);
  float rs = rsqrtf(var + 1e-5f);
  float4 gv = *(const float4*)(g + L * 4);
  float4 bv = *(const float4*)(b + L * 4);
  float4 out;
  out.x = d0 * rs * gv.x + bv.x;
  out.y = d1 * rs * gv.y + bv.y;
  out.z = d2 * rs * gv.z + bv.z;
  out.w = d3 * rs * gv.w + bv.w;
  *(float4*)(Y + (size_t)row * 128 + L * 4) = out;
}

// ---------------------------------------------------------------------------
// Fused FFN + LN:  out = LN( relu(X2 @ Wf1 + bf1) @ Wf2 + bf2 + X2 ).
// Block processes 32 rows; hidden 32x512 tile stays in LDS (never hits HBM).
// Phase 1 (K=128): each wave owns col tiles (wave + 8t), B fragment preloaded
//   to registers and reused across both row halves.
// Phase 2 (K=512): wave owns one 16-col tile; each B load pair feeds the
//   WMMAs of both row halves (1 VMEM load per WMMA).
// ---------------------------------------------------------------------------
__global__ __launch_bounds__(256) void ffn_ln_kernel(
    const float* __restrict__ X2, const float* __restrict__ Wf1,
    const float* __restrict__ bf1, const float* __restrict__ Wf2,
    const float* __restrict__ bf2, const float* __restrict__ g2,
    const float* __restrict__ b2, float* __restrict__ out, int M) {
  __shared__ float Xs[32][132];  // input tile (kept for residual)
  __shared__ float Hs[32][516];  // hidden tile (relu'd)
  __shared__ float Os[32][132];  // out3 tile

  const int tid = threadIdx.x;
  const int wave = tid >> 5, lane = tid & 31;
  const int rowBase = blockIdx.x * 32;

  // Stage X2 tile.
  {
    int r = tid >> 3;
    int c0 = (tid & 7) * 16;
    int rr = rowBase + r;
    if (rr > M - 1) rr = M - 1;
    const float* src = X2 + (size_t)rr * 128 + c0;
    #pragma unroll
    for (int i = 0; i < 4; ++i) {
      float4 t = *(const float4*)(src + 4 * i);
      Xs[r][c0 + 4 * i + 0] = t.x;
      Xs[r][c0 + 4 * i + 1] = t.y;
      Xs[r][c0 + 4 * i + 2] = t.z;
      Xs[r][c0 + 4 * i + 3] = t.w;
    }
  }
  __syncthreads();

  const int m = lane & 15;
  const int kb_off = (lane >> 4) << 1;
  const int hi8 = (lane >> 4) << 3;

  // Phase 1: H = relu(Xs @ Wf1 + bf1), 512 cols. 4 col tiles per wave.
  for (int t = 0; t < 4; ++t) {
    int col = (wave + 8 * t) * 16 + m;

    const float* Wb = Wf1 + col + (size_t)kb_off * 512;
    v2f Bf[32];
    #pragma unroll
    for (int i = 0; i < 32; ++i) {
      Bf[i].x = Wb[(size_t)(4 * i) * 512];
      Bf[i].y = Wb[(size_t)(4 * i + 1) * 512];
    }

    v8f acc0 = {}, acc1 = {};
    #pragma unroll
    for (int i = 0; i < 32; ++i) {
      int kb = 4 * i + kb_off;
      v2f a0; a0.x = Xs[m][kb];      a0.y = Xs[m][kb + 1];
      v2f a1; a1.x = Xs[16 + m][kb]; a1.y = Xs[16 + m][kb + 1];
      acc0 = wmma_f32x4(a0, Bf[i], acc0);
      acc1 = wmma_f32x4(a1, Bf[i], acc1);
    }

    float bb = bf1[col];
    #pragma unroll
    for (int j = 0; j < 8; ++j) {
      Hs[j + hi8][col]      = fmaxf(acc0[j] + bb, 0.0f);
      Hs[16 + j + hi8][col] = fmaxf(acc1[j] + bb, 0.0f);
    }
  }
  __syncthreads();

  // Phase 2: O = Hs @ Wf2 + bf2, 128 cols (one 16-col tile per wave), K=512.
  {
    int col = wave * 16 + m;
    const float* Wb = Wf2 + col + (size_t)kb_off * 128;
    v8f acc0 = {}, acc1 = {};
    #pragma unroll 8
    for (int kk = 0; kk < 512; kk += 4) {
      int kb = kk + kb_off;
      v2f b; b.x = Wb[(size_t)kk * 128]; b.y = Wb[(size_t)(kk + 1) * 128];
      v2f a0; a0.x = Hs[m][kb];      a0.y = Hs[m][kb + 1];
      v2f a1; a1.x = Hs[16 + m][kb]; a1.y = Hs[16 + m][kb + 1];
      acc0 = wmma_f32x4(a0, b, acc0);
      acc1 = wmma_f32x4(a1, b, acc1);
    }
    float bb = bf2[col];
    #pragma unroll
    for (int j = 0; j < 8; ++j) {
      Os[j + hi8][col]      = acc0[j] + bb;
      Os[16 + j + hi8][col] = acc1[j] + bb;
    }
  }
  __syncthreads();

  // LN over (Os + Xs); each wave normalizes 4 rows.
  for (int i = 0; i < 4; ++i) {
    int r = wave * 4 + i;
    int row = rowBase + r;
    float t0 = Os[r][4 * lane + 0] + Xs[r][4 * lane + 0];
    float t1 = Os[r][4 * lane + 1] + Xs[r][4 * lane + 1];
    float t2 = Os[r][4 * lane + 2] + Xs[r][4 * lane + 2];
    float t3 = Os[r][4 * lane + 3] + Xs[r][4 * lane + 3];
    float mu = waveReduceSum(t0 + t1 + t2 + t3) * (1.0f / 128.0f);
    float d0 = t0 - mu, d1 = t1 - mu, d2 = t2 - mu, d3 = t3 - mu;
    float var = waveReduceSum(d0 * d0 + d1 * d1 + d2 * d2 + d3 * d3) * (1.0f / 128.0f);
    float rs = rsqrtf(var + 1e-5f);
    if (row < M) {
      float4 gv = *(const float4*)(g2 + lane * 4);
      float4 bv = *(const float4*)(b2 + lane * 4);
      float4 o;
      o.x = d0 * rs * gv.x + bv.x;
      o.y = d1 * rs * gv.y + bv.y;
      o.z = d2 * rs * gv.z + bv.z;
      o.w = d3 * rs * gv.w + bv.w;
      *(float4*)(out + (size_t)row * 128 + lane * 4) = o;
    }
  }
}

// ---------------------------------------------------------------------------
// Host-side launch.
// ---------------------------------------------------------------------------
extern "C" void kernel_launch(void* const* d_in, const int* in_sizes, int n_in,
                              void* d_out, int out_size, void* d_ws, size_t ws_size,
                              hipStream_t stream) {
  const float* x = (const float*)d_in[0];
  const long long* ei = (const long long*)d_in[1];  // (2,E) int64: row0=src,row1=dst
  const float* Wq = (const float*)d_in[2];
  const float* bq = (const float*)d_in[3];
  const float* Wk = (const float*)d_in[4];
  const float* bk = (const float*)d_in[5];
  const float* Wv = (const float*)d_in[6];
  const float* bv = (const float*)d_in[7];
  const float* Wskip = (const float*)d_in[8];
  const float* bskip = (const float*)d_in[9];
  const float* Wo = (const float*)d_in[10];
  const float* bo = (const float*)d_in[11];
  const float* g1 = (const float*)d_in[12];
  const float* b1 = (const float*)d_in[13];
  const float* Wf1 = (const float*)d_in[14];
  const float* bf1 = (const float*)d_in[15];
  const float* Wf2 = (const float*)d_in[16];
  const float* bf2 = (const float*)d_in[17];
  const float* g2 = (const float*)d_in[18];
  const float* b2 = (const float*)d_in[19];

  const int N = in_sizes[0] / 128;
  const int E = in_sizes[1] / 2;

  // Workspace layout (floats). Total ~203 MB.
  float* ws = (float*)d_ws;
  size_t off = 0;
  float* Wcat = ws + off; off += (size_t)128 * 512;
  float* bcat = ws + off; off += 512;
  float* qkvs = ws + off; off += (size_t)N * 512;  // q|k|v|skip per node
  float* alpha = ws + off; off += (size_t)E * 8;   // logits -> exp values
  float* maxa = ws + off; off += (size_t)N * 8;
  float* den  = ws + off; off += (size_t)N * 8;
  float* conv = ws + off; off += (size_t)N * 128;  // skip + attention agg
  float* out1 = ws + off; off += (size_t)N * 128;
  float* out2 = ws + off; off += (size_t)N * 128;

  const int rowBlocks64 = (N + 63) / 64;
  const int rowBlocks32 = (N + 31) / 32;

  // 1) Pack fused QKV+skip weights.
  pack_w_kernel<<<(128 * 512 + 255) / 256, 256, 0, stream>>>(
      Wq, Wk, Wv, Wskip, bq, bk, bv, bskip, Wcat, bcat);

  // 2) qkvs = x @ Wcat + bcat   (N x 128 x 512, WMMA; 128-col block stripes)
  gemm128_bias_kernel<512><<<dim3(rowBlocks64, 4), 256, 0, stream>>>(
      x, Wcat, bcat, qkvs, N);

  // 3) Init softmax stats; conv initialized with skip projection.
  init_buffers_kernel<<<((size_t)N * 128 + 255) / 256, 256, 0, stream>>>(
      maxa, den, qkvs, conv, N);

  // 4) Edge attention.
  edge_logits_kernel<<<((size_t)E * 8 + 255) / 256, 256, 0, stream>>>(
      qkvs, ei, alpha, maxa, E);
  edge_expden_kernel<<<((size_t)E * 8 + 255) / 256, 256, 0, stream>>>(
      alpha, ei, maxa, den, E);
  edge_aggregate_kernel<<<((size_t)E * 32 + 255) / 256, 256, 0, stream>>>(
      qkvs, ei, alpha, den, conv, E);

  // 5) out1 = conv @ Wo + bo   (WMMA)
  gemm128_bias_kernel<128><<<dim3(rowBlocks64, 1), 256, 0, stream>>>(
      conv, Wo, bo, out1, N);

  // 6) out2 = LN(out1 + x)
  ln_residual_kernel<<<(N + 7) / 8, 256, 0, stream>>>(out1, x, g1, b1, out2, N);

  // 7) d_out = LN(relu(out2@Wf1+bf1)@Wf2+bf2 + out2)  (fused WMMA FFN)
  ffn_ln_kernel<<<rowBlocks32, 256, 0, stream>>>(
      out2, Wf1, bf1, Wf2, bf2, g2, b2, (float*)d_out, N);
}